// RBFW_57595511439583
// MI455X (gfx1250) — compile-verified
//
#include <hip/hip_runtime.h>

// Problem constants (match reference: N=320, D=64)
constexpr int NN = 320;
constexpr int DD = 64;
constexpr int NTILE = NN / 16;       // 20 tiles per output dim
constexpr int LSTR = DD + 1;         // LDS row stride (bank-conflict padding)

typedef __attribute__((ext_vector_type(2))) float v2f;
typedef __attribute__((ext_vector_type(8))) float v8f;

__global__ __launch_bounds__(256) void RBFW_stein_kernel(
    const float* __restrict__ x,      // [N, D]
    const float* __restrict__ sigp,   // scalar
    const float* __restrict__ score,  // [N, D]
    const float* __restrict__ v,      // [N, D]
    const float* __restrict__ gvs,    // [N, D]
    float* __restrict__ out)
{
    // LDS panels for row i of the batched GEMMs (per-workgroup i)
    __shared__ float sOD[NN * LSTR];  // od[i, j, d]
    __shared__ float sKD[NN * LSTR];  // k_d[i, j, d]
    __shared__ float sAM[NN * LSTR];  // raw A = (1 - g*od^2) * score[j,d] * kd * od
    __shared__ float sSQ[NN];         // sum_d od^2      (per j)
    __shared__ float sKK[NN];         // k[i, j]         (per j)

    const int i    = blockIdx.x;
    const int tid  = threadIdx.x;
    const int wave = tid >> 5;
    const int lane = tid & 31;

    const float sigma = sigp[0];
    const float gamma = 1.0f / (sigma * sigma + 1e-8f);

    // Output section bases (tuple concatenated flat in return order)
    float* out_k   = out;                                   // [N,N]
    float* out_kxj = out_k   + (size_t)NN * NN;             // [N,N,D]
    float* out_kxi = out_kxj + (size_t)NN * NN * DD;        // [N,N,D]
    float* out_tr  = out_kxi + (size_t)NN * NN * DD;        // [N,N]
    float* out_s5  = out_tr  + (size_t)NN * NN;             // [N,N,N] score_t_k_xk_xi_k_xi
    float* out_s6  = out_s5  + (size_t)NN * NN * NN;        // [N,N,N] tr_k_xj_xi_t_k_xk_xi
    float* out_vt  = out_s6  + (size_t)NN * NN * NN;        // [N,N]

    // Per-lane slices of row i (d = lane and lane+32)
    const float xi0 = x  [(size_t)i * DD + lane];
    const float xi1 = x  [(size_t)i * DD + lane + 32];
    const float vi0 = v  [(size_t)i * DD + lane];
    const float vi1 = v  [(size_t)i * DD + lane + 32];
    const float gi0 = gvs[(size_t)i * DD + lane];
    const float gi1 = gvs[(size_t)i * DD + lane + 32];

    // vdot_i = sum_d v[i,d] * gvs[i,d]  (wave32 butterfly)
    float vdot = vi0 * gi0 + vi1 * gi1;
    #pragma unroll
    for (int s = 16; s > 0; s >>= 1) vdot += __shfl_xor(vdot, s, 32);

    // ---------------- Phase 1: per-row elementwise + reductions ----------------
    for (int j = wave; j < NN; j += 8) {
        const float xj0 = x[(size_t)j * DD + lane];
        const float xj1 = x[(size_t)j * DD + lane + 32];
        const float sc0 = score[(size_t)j * DD + lane];
        const float sc1 = score[(size_t)j * DD + lane + 32];

        const float od0 = xi0 - xj0;
        const float od1 = xi1 - xj1;
        const float g0  = gamma * od0 * od0;   // gods
        const float g1  = gamma * od1 * od1;
        const float kd0 = __expf(-0.5f * g0);
        const float kd1 = __expf(-0.5f * g1);

        sOD[j * LSTR + lane]      = od0;
        sOD[j * LSTR + lane + 32] = od1;
        sKD[j * LSTR + lane]      = kd0;
        sKD[j * LSTR + lane + 32] = kd1;
        // raw A (final scale -(gamma^2)/(D*D) folded into epilogue)
        sAM[j * LSTR + lane]      = (1.0f - g0) * sc0 * kd0 * od0;
        sAM[j * LSTR + lane + 32] = (1.0f - g1) * sc1 * kd1 * od1;

        // k_xj = gamma * k_d * od / D ; k_xi = -k_xj
        const float kxj0 = (gamma / (float)DD) * kd0 * od0;
        const float kxj1 = (gamma / (float)DD) * kd1 * od1;
        const size_t pb = ((size_t)i * NN + j) * DD;
        out_kxj[pb + lane]      =  kxj0;
        out_kxj[pb + lane + 32] =  kxj1;
        out_kxi[pb + lane]      = -kxj0;
        out_kxi[pb + lane + 32] = -kxj1;

        // Row reductions (over d)
        float s_kd = kd0 + kd1;                               // -> k
        float s_sq = od0 * od0 + od1 * od1;                   // -> sq
        float s_tr = (1.0f - g0) * kd0 + (1.0f - g1) * kd1;   // -> tr
        float s_a  = vi0 * od0 + vi1 * od1;                   // v_i . od
        float s_b  = gi0 * od0 + gi1 * od1;                   // gvs_i . od
        #pragma unroll
        for (int s = 16; s > 0; s >>= 1) {
            s_kd += __shfl_xor(s_kd, s, 32);
            s_sq += __shfl_xor(s_sq, s, 32);
            s_tr += __shfl_xor(s_tr, s, 32);
            s_a  += __shfl_xor(s_a,  s, 32);
            s_b  += __shfl_xor(s_b,  s, 32);
        }
        if (lane == 0) {
            const float kij = s_kd * (1.0f / (float)DD);
            const size_t o = (size_t)i * NN + j;
            out_k [o] = kij;
            sKK[j]    = kij;
            sSQ[j]    = s_sq;
            out_tr[o] = gamma * s_tr * (1.0f / (float)DD);
            out_vt[o] = (-2.0f * gamma * s_a * s_b + vdot) * kij * 2.0f * gamma;
        }
    }
    __syncthreads();

    // ---------------- Phase 2: WMMA tiles over the [N,N] plane for this i ------
    const int   m      = lane & 15;        // A-frag row / B-frag col within tile
    const int   koff   = (lane >> 4) * 2;  // K sub-offset per ISA 16x4 f32 layout
    const float c5     = -(gamma * gamma) / (float)(DD * DD); // -(g^2/4096)
    const float fourg2 = 4.0f * gamma * gamma;

    for (int t = wave; t < NTILE * NTILE; t += 8) {
        const int j0 = (t / NTILE) * 16;   // output rows (j)
        const int c0 = (t % NTILE) * 16;   // output cols (k)

        v8f acc1 = {0.f, 0.f, 0.f, 0.f, 0.f, 0.f, 0.f, 0.f}; // A . kd^T
        v8f acc2 = {0.f, 0.f, 0.f, 0.f, 0.f, 0.f, 0.f, 0.f}; // od . od^T

        const float* rA2 = &sOD[(j0 + m) * LSTR + koff];
        const float* rB2 = &sOD[(c0 + m) * LSTR + koff];
        const float* rA1 = &sAM[(j0 + m) * LSTR + koff];
        const float* rB1 = &sKD[(c0 + m) * LSTR + koff];

        #pragma unroll
        for (int k0 = 0; k0 < DD; k0 += 4) {
            v2f a2 = { rA2[k0], rA2[k0 + 1] };
            v2f b2 = { rB2[k0], rB2[k0 + 1] };
            acc2 = __builtin_amdgcn_wmma_f32_16x16x4_f32(
                false, a2, false, b2, (short)0, acc2, false, false);

            v2f a1 = { rA1[k0], rA1[k0 + 1] };
            v2f b1 = { rB1[k0], rB1[k0 + 1] };
            acc1 = __builtin_amdgcn_wmma_f32_16x16x4_f32(
                false, a1, false, b1, (short)0, acc1, false, false);
        }

        // Epilogue + store (C/D layout: VGPR r -> row j0 + r + 8*(lane>>4), col = c0 + m)
        const int   row0 = j0 + (lane >> 4) * 8;
        const int   col  = c0 + m;
        const size_t base = (size_t)i * NN * NN + col;
        const float sqc = sSQ[col];
        const float kkc = sKK[col];
        #pragma unroll
        for (int r = 0; r < 8; ++r) {
            const int row = row0 + r;
            const size_t o = base + (size_t)row * NN;
            out_s5[o] = acc1[r] * c5;
            const float dot = acc2[r];
            const float t1  = fourg2 * dot * dot;
            const float t2  = -2.0f * gamma * (sSQ[row] + sqc);
            out_s6[o] = (t1 + t2 + (float)DD) * sKK[row] * kkc * fourg2;
        }
    }
}

extern "C" void kernel_launch(void* const* d_in, const int* in_sizes, int n_in,
                              void* d_out, int out_size, void* d_ws, size_t ws_size,
                              hipStream_t stream) {
    const float* x     = (const float*)d_in[0];
    const float* sigma = (const float*)d_in[1];
    const float* score = (const float*)d_in[2];
    const float* vv    = (const float*)d_in[3];
    const float* gvs   = (const float*)d_in[4];
    float* out = (float*)d_out;
    (void)in_sizes; (void)n_in; (void)out_size; (void)d_ws; (void)ws_size;

    RBFW_stein_kernel<<<dim3(NN), dim3(256), 0, stream>>>(x, sigma, score, vv, gvs, out);
}